// KANLayer_19859928777110
// MI455X (gfx1250) — compile-verified
//
#include <hip/hip_runtime.h>

// Problem dims
#define BB 32768
#define II 256
#define OO 256
#define NBD 256
#define HH 16

// Tiling
#define BM 128                    // batch rows per block
#define KB 32                     // k-chunk (input dim) per iteration
#define NCH (II / KB)             // 8 chunks
#define XS 36                     // LDS row stride (floats): 32 + 4 pad (TDM pad cfg 4/3)
#define XT_FLOATS (BM * XS)       // 4608 : x/phi tile, rows = batch m, cols = k
#define WT_FLOATS (OO * XS)       // 9216 : wsum tile, rows = output o, cols = k
#define LDS_FLOATS (2 * XT_FLOATS + 2 * WT_FLOATS)   // 27648 floats
#define LDS_BYTES (LDS_FLOATS * 4)                   // 110592 B (dynamic LDS; 320KB/WGP)

typedef __attribute__((ext_vector_type(2))) float v2f;
typedef __attribute__((ext_vector_type(8))) float v8f;
typedef __attribute__((ext_vector_type(4))) unsigned int v4u;
typedef __attribute__((ext_vector_type(4))) int v4i;
typedef __attribute__((ext_vector_type(8))) int v8i;

// group1 word0: data_size=2 (4B) | pad_enable | pad_interval=4 (32 DW) | pad_amount=3 (4 DW)
// -> LDS row stride 36 floats for 32-float tile rows.
#define TCFG ((2u << 16) | (1u << 20) | (4u << 22) | (3u << 25))

// ---------------------------------------------------------------------------
// TDM: DMA a 2D tile [tile1 rows x tile0 elems] of f32 from global -> LDS,
// inserting LDS padding per `cfg`. Issued once by the executing wave;
// completion tracked on TENSORcnt. Descriptor layout per CDNA5 ISA §8.3/8.4.
// ---------------------------------------------------------------------------
__device__ __forceinline__ void tdm_load_2d(unsigned lds_addr, const void* gptr,
                                            unsigned cfg, unsigned dim0, unsigned dim1,
                                            unsigned tile0, unsigned tile1,
                                            unsigned stride0) {
  const unsigned long long ga = (unsigned long long)(size_t)gptr;
  v4u g0;
  g0.x = 1u;                                               // count=1 (valid), user mode
  g0.y = lds_addr;                                         // LDS byte address
  g0.z = (unsigned)ga;                                     // global_addr[31:0]
  g0.w = (unsigned)((ga >> 32) & 0x1FFFFFFull) | (2u << 30);  // global_addr[56:32] | type=2
  v8i g1;
  g1[0] = (int)cfg;                                        // mask=0, data_size, pad cfg
  g1[1] = (int)((dim0 & 0xFFFFu) << 16);                   // tensor_dim0[15:0] (atomic addr = 0)
  g1[2] = (int)(((dim0 >> 16) & 0xFFFFu) | ((dim1 & 0xFFFFu) << 16));  // dim0 hi | dim1 lo
  g1[3] = (int)(((dim1 >> 16) & 0xFFFFu) | ((tile0 & 0xFFFFu) << 16)); // dim1 hi | tile_dim0
  g1[4] = (int)(tile1 & 0xFFFFu);                          // tile_dim1 | tile_dim2=0
  g1[5] = (int)stride0;                                    // tensor_dim0_stride[31:0]
  g1[6] = 0;                                               // stride0 hi | stride1 lo
  g1[7] = 0;
  v4i z4 = {0, 0, 0, 0};
#if __clang_major__ >= 23
  v8i z8 = {0, 0, 0, 0, 0, 0, 0, 0};
  __builtin_amdgcn_tensor_load_to_lds(g0, g1, z4, z4, z8, 0);   // amdgpu-toolchain (clang-23) arity
#else
  __builtin_amdgcn_tensor_load_to_lds(g0, g1, z4, z4, 0);       // ROCm 7.2 (clang-22) arity
#endif
}

__device__ __forceinline__ unsigned lds_byte_addr(const float* p) {
  // Low 32 bits of the generic address are the wave-relative LDS byte offset.
  return (unsigned)(size_t)(const void*)p;
}

// ---------------------------------------------------------------------------
// Kernel 1: wsum[o][i] = sum_k weights[o][i][k]   (67 MB streamed once)
// One wave per (o,i) row: 256 contiguous floats = 2x float4 per lane.
// ---------------------------------------------------------------------------
__global__ __launch_bounds__(256) void kan_wsum_kernel(const float* __restrict__ w,
                                                       float* __restrict__ wsum) {
  const int wid  = (blockIdx.x * 256 + threadIdx.x) >> 5;  // flat (o,i) in 0 .. O*I-1
  const int lane = threadIdx.x & 31;
  const float* row = w + (size_t)wid * NBD;
  const float4 a = *(const float4*)(row + lane * 8);
  const float4 b = *(const float4*)(row + lane * 8 + 4);
  float s = ((a.x + a.y) + (a.z + a.w)) + ((b.x + b.y) + (b.z + b.w));
#pragma unroll
  for (int off = 16; off > 0; off >>= 1) s += __shfl_down(s, off, 32);
  if (lane == 0) wsum[wid] = s;   // same [O][I] layout as the weights' leading dims
}

// ---------------------------------------------------------------------------
// Kernel 2: fused phi + GEMM, TDM double-buffered.
// Block = 256 threads (8 waves), computes out[row0:row0+128, 0:256].
// ---------------------------------------------------------------------------
__global__ __launch_bounds__(256) void kan_fused_kernel(
    const float* __restrict__ x, const float* __restrict__ w1,
    const float* __restrict__ b1, const float* __restrict__ w2,
    const float* __restrict__ b2, const float* __restrict__ wsum,
    float* __restrict__ out) {
  extern __shared__ float lds[];

  const int tid  = threadIdx.x;
  const int lane = tid & 31;
  const int wave = tid >> 5;
  const int half = lane >> 4;
  const int l16  = lane & 15;
  const int row0 = blockIdx.x * BM;

  v8f acc[16];
#pragma unroll
  for (int n = 0; n < 16; ++n) acc[n] = {};

  // Prologue: DMA chunk 0 (wave 0 only; TDM ignores EXEC, one issue per wave).
  if (tid < 32) {
    // x tile: BM rows x KB cols from x[row0:, 0:KB]
    tdm_load_2d(lds_byte_addr(lds), x + (size_t)row0 * II,
                TCFG, II, BB, KB, BM, II);
    // wsum tile: OO rows x KB cols from wsum[0:, 0:KB]
    tdm_load_2d(lds_byte_addr(lds + 2 * XT_FLOATS), wsum,
                TCFG, II, OO, KB, OO, II);
  }

  for (int c = 0; c < NCH; ++c) {
    const int k0  = c * KB;
    const int cur = c & 1;
    float* xt = lds + cur * XT_FLOATS;                    // [BM][XS]: phi rows m, cols k
    float* wt = lds + 2 * XT_FLOATS + cur * WT_FLOATS;    // [OO][XS]: wsum rows o, cols k

    if (tid < 32) {
      if (c + 1 < NCH) {
        const int k1  = k0 + KB;
        const int nxt = (c + 1) & 1;
        // Buffer `nxt` was last read in chunk c-1; the end-of-iter barrier
        // already guaranteed those reads completed.
        tdm_load_2d(lds_byte_addr(lds + nxt * XT_FLOATS),
                    x + (size_t)row0 * II + k1, TCFG, II, BB, KB, BM, II);
        tdm_load_2d(lds_byte_addr(lds + 2 * XT_FLOATS + nxt * WT_FLOATS),
                    wsum + k1, TCFG, II, OO, KB, OO, II);
        __builtin_amdgcn_s_wait_tensorcnt((short)2);  // in-order: chunk c's pair done
      } else {
        __builtin_amdgcn_s_wait_tensorcnt((short)0);
      }
    }
    __syncthreads();  // publish chunk c tiles to all waves

    // ---- phi in place on xt: thread owns column kk=lane (i=k0+lane),
    //      rows wave*16 .. wave*16+15 (exactly the rows this wave's WMMA reads).
    {
      const int i  = k0 + lane;
      const int mb = wave * 16;
      float xv[16], pa[16];
#pragma unroll
      for (int j = 0; j < 16; ++j) xv[j] = xt[(mb + j) * XS + lane];
      const float bias2 = b2[i];
#pragma unroll
      for (int j = 0; j < 16; ++j) pa[j] = bias2;
#pragma unroll
      for (int hb = 0; hb < HH; hb += 4) {
        const float4 wv = *(const float4*)&w1[i * HH + hb];
        const float4 bv = *(const float4*)&b1[i * HH + hb];
        const float4 vv = *(const float4*)&w2[i * HH + hb];
        const float wa[4] = {wv.x, wv.y, wv.z, wv.w};
        const float ba[4] = {bv.x, bv.y, bv.z, bv.w};
        const float va[4] = {vv.x, vv.y, vv.z, vv.w};
#pragma unroll
        for (int hh = 0; hh < 4; ++hh) {
#pragma unroll
          for (int j = 0; j < 16; ++j) {
            const float z = fmaf(xv[j], wa[hh], ba[hh]);
            const float e = __expf(-z);                        // v_exp_f32
            const float sg = __builtin_amdgcn_rcpf(1.0f + e);  // v_rcp_f32
            pa[j] = fmaf(z * sg, va[hh], pa[j]);               // silu(z)*w2 + acc
          }
        }
      }
#pragma unroll
      for (int j = 0; j < 16; ++j) xt[(mb + j) * XS + lane] = pa[j];
    }
    // No barrier needed: this wave's A-fragment rows were written by this wave,
    // and DS operations from one wave execute in order.

    // ---- WMMA: D += A(16x4) x B(4x16), f32.
    // A layout (ISA 7.12.2): VGPR v, lane-half h -> K = v + 2h, M = lane%16.
    // B symmetric:           VGPR v, lane-half h -> K = v + 2h, N = lane%16.
    // Both tiles are [row][k] with stride XS, so every fragment is one
    // aligned ds_load_b64 of two consecutive k values (conflict-free banks).
    const int arow = (wave * 16 + l16) * XS;
    const int brow = l16 * XS;
#pragma unroll
    for (int cc = 0; cc < KB / 4; ++cc) {
      const int kb = cc * 4 + 2 * half;
      v2f a = *(const v2f*)&xt[arow + kb];                 // K = kb, kb+1
#pragma unroll
      for (int n = 0; n < 16; ++n) {
        v2f b = *(const v2f*)&wt[brow + n * 16 * XS + kb]; // o = n*16+l16, K = kb, kb+1
        acc[n] = __builtin_amdgcn_wmma_f32_16x16x4_f32(
            false, a, false, b, (short)0, acc[n], false, false);
      }
    }
    __syncthreads();  // all reads of chunk c done before its buffer is re-DMA'd
  }

  // ---- Store: C/D layout VGPR j -> M = j + 8*half, N = lane%16 ----
  const int mrow = row0 + wave * 16 + 8 * half;
#pragma unroll
  for (int n = 0; n < 16; ++n) {
#pragma unroll
    for (int j = 0; j < 8; ++j) {
      out[(size_t)(mrow + j) * OO + n * 16 + l16] = acc[n][j];
    }
  }
}

// ---------------------------------------------------------------------------
extern "C" void kernel_launch(void* const* d_in, const int* in_sizes, int n_in,
                              void* d_out, int out_size, void* d_ws, size_t ws_size,
                              hipStream_t stream) {
  const float* x  = (const float*)d_in[0];   // [B, I]
  const float* w  = (const float*)d_in[1];   // [O, I, NB]
  const float* w1 = (const float*)d_in[2];   // [I, H]
  const float* b1 = (const float*)d_in[3];   // [I, H]
  const float* w2 = (const float*)d_in[4];   // [I, H]
  const float* b2 = (const float*)d_in[5];   // [I]
  float* out = (float*)d_out;                // [B, O]
  float* wsum = (float*)d_ws;                // [O, I] scratch (256 KB)

  kan_wsum_kernel<<<(OO * II) / 8, 256, 0, stream>>>(w, wsum);
  kan_fused_kernel<<<BB / BM, 256, LDS_BYTES, stream>>>(x, w1, b1, w2, b2, wsum, out);
}